// ModelNew_10522669875251
// MI455X (gfx1250) — compile-verified
//
#include <hip/hip_runtime.h>

// CSR multi-head SpMM via V_WMMA_F32_16X16X4_F32 segment-sum on gfx1250.
//
// out[r, h, d] = sum_{e in [row_ptr[r], row_ptr[r+1])} ew[e,h] * nf[col[e], h, d]
//
// One wave32 owns a block of 16 consecutive destination rows; CSR makes that
// block's edge set one contiguous range [row_ptr[r0], row_ptr[r0+16]).
// Walk it K=4 edges at a time:
//   A[16x4]  = {0,1} row-membership indicators (does row m own edge base+k ?)
//   B[4x16]  = messages w[e,h]*f[col[e],h,d]; accumulator g covers output
//              columns hd = 4n + g  (n = D-tile column). This grouping makes
//              each lane's 4 feature values contiguous (one b128 load = the
//              edge's full 256B feature row per half-wave) and makes the head
//              index h = (4n+g)>>3 = n>>1 independent of g (one weight scalar
//              per lane per edge).
//   C[16x16] = f32 accumulator tiles -> 8 coalesced b128 stores per wave.
//
// VGPR layouts per CDNA5 ISA 7.12.2 (wave32):
//   A 16x4 f32 : lane L (m=L&15, hi=L>>4): VGPR0 -> K=2*hi, VGPR1 -> K=2*hi+1
//   B 4x16 f32 : lane L (n=L&15):          VGPR0 -> K=2*hi, VGPR1 -> K=2*hi+1
//   C/D 16x16  : VGPR j, lanes 0-15 -> (M=j, N=lane); lanes 16-31 -> (M=j+8, N=lane-16)

typedef float v2f __attribute__((ext_vector_type(2)));
typedef float v4f __attribute__((ext_vector_type(4)));
typedef float v8f __attribute__((ext_vector_type(8)));

#define HEADS 8
#define HD    64   // HEADS * HEAD_DIM
#define ROWS_PER_WAVE 16

__device__ __forceinline__ int imin(int a, int b) { return a < b ? a : b; }

__global__ __launch_bounds__(256) void spmm_wmma16x16x4(
    const int*   __restrict__ row_ptr,   // [N+1]
    const int*   __restrict__ col_idx,   // [E]
    const float* __restrict__ ew,        // [E, 8]
    const float* __restrict__ nf,        // [N, 64]
    float*       __restrict__ out,       // [N, 64]
    int N, int E)
{
  const int lane   = threadIdx.x & 31;
  const int waveId = threadIdx.x >> 5;
  const int blk    = blockIdx.x * (blockDim.x >> 5) + waveId; // 16-row block id
  const int r0     = blk * ROWS_PER_WAVE;
  if (r0 >= N) return;                  // wave-uniform: EXEC stays full below

  const int n  = lane & 15;             // D-tile column / A-tile row index
  const int hi = lane >> 4;             // half-wave -> K slots {2hi, 2hi+1}

  // Per-lane row-ownership bounds for indicator matrix A. Clamped rows >= N
  // only feed D rows that are never stored.
  const int rr    = imin(r0 + n, N - 1);
  const int start = row_ptr[rr];
  const int end   = (r0 + n < N) ? row_ptr[rr + 1] : row_ptr[rr];

  const int blockStart = row_ptr[r0];
  const int blockEnd   = row_ptr[imin(r0 + ROWS_PER_WAVE, N)];

  const int wOff = n >> 1;              // head index for all 4 of this lane's columns

  v8f acc0 = {}, acc1 = {}, acc2 = {}, acc3 = {};

#pragma unroll 2
  for (int base = blockStart; base < blockEnd; base += 4) {
    const int e0 = base + 2 * hi;       // this half-wave's two K slots
    const int e1 = e0 + 1;

    // Indicator A: 1.0 iff the edge belongs to row n of this block. Edges past
    // blockEnd / E fail e < end, so clamped garbage is multiplied by zero.
    const float a0 = (e0 >= start && e0 < end) ? 1.0f : 0.0f;
    const float a1 = (e1 >= start && e1 < end) ? 1.0f : 0.0f;
    const v2f A = {a0, a1};

    // Clamp so no out-of-bounds address is ever formed.
    const int ec0 = imin(e0, E - 1);
    const int ec1 = imin(e1, E - 1);
    const int c0 = col_idx[ec0];
    const int c1 = col_idx[ec1];

    // One b128 per edge: lanes 0-15 of a half-wave fetch the edge's entire
    // 256B feature row, perfectly coalesced out of L2.
    const v4f fv0 = *(const v4f*)(nf + (size_t)c0 * HD + 4 * n);
    const v4f fv1 = *(const v4f*)(nf + (size_t)c1 * HD + 4 * n);
    // One weight scalar per edge per lane (h = n>>1 for all 4 columns).
    const float wv0 = ew[(size_t)ec0 * HEADS + wOff];
    const float wv1 = ew[(size_t)ec1 * HEADS + wOff];

    const v4f m0 = fv0 * wv0;           // messages for edge e0, columns 4n..4n+3
    const v4f m1 = fv1 * wv1;           // messages for edge e1

    acc0 = __builtin_amdgcn_wmma_f32_16x16x4_f32(false, A, false, (v2f){m0.x, m1.x},
                                                 (short)0, acc0, false, false);
    acc1 = __builtin_amdgcn_wmma_f32_16x16x4_f32(false, A, false, (v2f){m0.y, m1.y},
                                                 (short)0, acc1, false, false);
    acc2 = __builtin_amdgcn_wmma_f32_16x16x4_f32(false, A, false, (v2f){m0.z, m1.z},
                                                 (short)0, acc2, false, false);
    acc3 = __builtin_amdgcn_wmma_f32_16x16x4_f32(false, A, false, (v2f){m0.w, m1.w},
                                                 (short)0, acc3, false, false);
  }

  // Store: D VGPR j, lanes 0-15 -> row r0+j, lanes 16-31 -> row r0+j+8.
  // {acc0[j],acc1[j],acc2[j],acc3[j]} are output columns 4n..4n+3 -> one b128
  // per j; 16 lanes cover the row's full 256B, coalesced.
  const int mBase = r0 + 8 * hi;
  if (r0 + ROWS_PER_WAVE <= N) {        // wave-uniform fast path (always, here)
#pragma unroll
    for (int j = 0; j < 8; ++j) {
      v4f v = {acc0[j], acc1[j], acc2[j], acc3[j]};
      *(v4f*)(out + (size_t)(mBase + j) * HD + 4 * n) = v;
    }
  } else {
#pragma unroll
    for (int j = 0; j < 8; ++j) {
      if (mBase + j < N) {
        v4f v = {acc0[j], acc1[j], acc2[j], acc3[j]};
        *(v4f*)(out + (size_t)(mBase + j) * HD + 4 * n) = v;
      }
    }
  }
}

extern "C" void kernel_launch(void* const* d_in, const int* in_sizes, int n_in,
                              void* d_out, int out_size, void* d_ws, size_t ws_size,
                              hipStream_t stream) {
  const int*   row_ptr = (const int*)  d_in[0];   // [N+1]
  const int*   col_idx = (const int*)  d_in[1];   // [E]
  const float* ew      = (const float*)d_in[2];   // [E, 8]
  const float* nf      = (const float*)d_in[3];   // [N, 64]
  float*       out     = (float*)      d_out;     // [N, 64]

  const int N = in_sizes[0] - 1;
  const int E = in_sizes[1];

  const int nBlk       = (N + ROWS_PER_WAVE - 1) / ROWS_PER_WAVE; // 3125
  const int wavesPerWG = 256 / 32;
  const int grid       = (nBlk + wavesPerWG - 1) / wavesPerWG;

  spmm_wmma16x16x4<<<grid, 256, 0, stream>>>(row_ptr, col_idx, ew, nf, out, N, E);
}